// DibxBase_7756710937000
// MI455X (gfx1250) — compile-verified
//
#include <hip/hip_runtime.h>
#include <hip/hip_bf16.h>
#include <math.h>
#include <stdint.h>

typedef float v2f __attribute__((ext_vector_type(2)));
typedef float v8f __attribute__((ext_vector_type(8)));

#define NUNIT      48
#define GRAM_ELEMS 16384      // 128*128

// ---- d_out layout (floats), return order: logit, z, xt_loss, mp_loss ----
#define OUT_LOGIT  0
#define OUT_Z      1280
#define OUT_XT     (1280 + 25165824)
#define OUT_MP     (OUT_XT + 48)

// ---- workspace layout (floats) ----
#define WS_ZSMALL  0
#define WS_MASKED  1572864
#define WS_GRAM    (WS_MASKED + 25165824)          // 96 grams
#define WS_KMAT    (WS_GRAM + 96 * GRAM_ELEMS)     // 144 kernel matrices
#define WS_ENT     (WS_KMAT + 144 * GRAM_ELEMS)    // 144 entropies
#define WS_HPOOL   (WS_ENT + 192)                  // 128*16 pooled decoder feats

// ------------------------------------------------------------------
// Encoder conv (stride 4, 4x4, C=3 -> 3) + sigmoid + KLDiv mp_loss
// ------------------------------------------------------------------
__global__ __launch_bounds__(256)
void enc_kernel(const float* __restrict__ x, const float* __restrict__ prior,
                const float* __restrict__ enc_w, const float* __restrict__ enc_b,
                float* __restrict__ zsmall, float* __restrict__ out_mp)
{
    int idx = blockIdx.x * 256 + threadIdx.x;
    if (idx >= 1572864) return;
    int ox = idx & 63, oy = (idx >> 6) & 63;
    int tmp = idx >> 12; int dch = tmp % 3; int b = tmp / 3;
    float acc = enc_b[dch];
    const float* wbase = enc_w + dch * 48;
    #pragma unroll
    for (int ic = 0; ic < 3; ++ic) {
        const float* xb = x + (((size_t)(b * 3 + ic) * 256 + oy * 4) * 256) + ox * 4;
        #pragma unroll
        for (int ky = 0; ky < 4; ++ky)
            #pragma unroll
            for (int kx = 0; kx < 4; ++kx)
                acc += xb[ky * 256 + kx] * wbase[ic * 16 + ky * 4 + kx];
    }
    float sig = 1.f / (1.f + expf(-acc));
    zsmall[idx] = sig;
    float pr = prior[idx];
    out_mp[idx] = pr * (logf(pr) - logf(sig + 1e-7f));   // GAMMA = 1
}

// ------------------------------------------------------------------
// Nearest-neighbor upsample of z_small -> z, and masked = x * z
// ------------------------------------------------------------------
__global__ __launch_bounds__(256)
void upsample_mask_kernel(const float* __restrict__ x, const float* __restrict__ zsmall,
                          float* __restrict__ out_z, float* __restrict__ masked)
{
    int idx = blockIdx.x * 256 + threadIdx.x;
    if (idx >= 25165824) return;
    int xc = idx & 255; int yr = (idx >> 8) & 255;
    int tmp = idx >> 16; int ch = tmp % 3; int b = tmp / 3;
    float zs = zsmall[((size_t)(b * 3 + ch) * 64 + (yr >> 2)) * 64 + (xc >> 2)];
    out_z[idx] = zs;
    masked[idx] = x[idx] * zs;
}

// ------------------------------------------------------------------
// Gram kernel: per block, G = X X^T for X = [128, 4096] patch slice.
// blockIdx.x = unit*2 + which   (which: 0 = x, 1 = masked)
// fp32 WMMA (V_WMMA_F32_16X16X4_F32); K-panels double-buffered in LDS
// via GLOBAL_LOAD_ASYNC_TO_LDS_B128 (ASYNCcnt-tracked, in-order done).
// ------------------------------------------------------------------
__global__ __launch_bounds__(256)
void gram_kernel(const float* __restrict__ x, const float* __restrict__ masked,
                 float* __restrict__ ws_gram)
{
    __shared__ float Xs[2][128][132];   // double-buffered K-panels, padded rows
    int bid   = blockIdx.x;             // 0..95
    int unit  = bid >> 1;
    int which = bid & 1;
    int d     = unit / 16;
    int patch = unit % 16;
    int pi = patch >> 2, pj = patch & 3;
    const float* src = which ? masked : x;

    int tid   = threadIdx.x;
    int wave  = tid >> 5;
    int lane  = tid & 31;
    int lhalf = lane >> 4;              // 0: K+0/1  1: K+2/3
    int l16   = lane & 15;
    int m0    = wave << 4;              // this wave's M tile row

    v8f acc[8];
    #pragma unroll
    for (int n = 0; n < 8; ++n)
        #pragma unroll
        for (int i = 0; i < 8; ++i) acc[n][i] = 0.f;

    // issue one 128x128 panel as 16 async b128 loads per thread
    auto issue_panel = [&](int buf, int kc) {
        #pragma unroll 4
        for (int it = 0; it < 16; ++it) {
            int i  = tid + it * 256;        // 16B-chunk index 0..4095
            int b  = i >> 5;                // 32 chunks per row
            int kk = (i & 31) << 2;         // chunk start within K panel
            int k  = kc + kk;
            int r = k >> 6, c = k & 63;
            const float* gp = src +
                (((size_t)(b * 3 + d) * 256 + pi * 64 + r) * 256) + pj * 64 + c;
            uint32_t ldsb = (uint32_t)(uintptr_t)(&Xs[buf][b][kk]);
            asm volatile("global_load_async_to_lds_b128 %0, %1, off"
                         :: "v"(ldsb), "v"((uint64_t)(uintptr_t)gp)
                         : "memory");
        }
    };

    issue_panel(0, 0);
    for (int kc = 0; kc < 4096; kc += 128) {
        int cur = (kc >> 7) & 1;
        if (kc + 128 < 4096) {
            issue_panel(cur ^ 1, kc + 128);       // prefetch next panel
            asm volatile("s_wait_asynccnt 16" ::: "memory");  // current panel done
        } else {
            asm volatile("s_wait_asynccnt 0" ::: "memory");
        }
        __syncthreads();

        for (int k = 0; k < 128; k += 4) {
            int kb = k + (lhalf << 1);
            v2f a;
            a.x = Xs[cur][m0 + l16][kb];
            a.y = Xs[cur][m0 + l16][kb + 1];
            #pragma unroll
            for (int n = 0; n < 8; ++n) {
                int row = (n << 4) + l16;          // B[k][n] = X[n][k]
                v2f bf;
                bf.x = Xs[cur][row][kb];
                bf.y = Xs[cur][row][kb + 1];
                acc[n] = __builtin_amdgcn_wmma_f32_16x16x4_f32(
                    false, a, false, bf, (short)0, acc[n], false, false);
            }
        }
        __syncthreads();   // protects cur^1 buffer reuse next round
    }

    // store C: VGPR i -> M = m0 + i (+8 for lanes 16-31), N = n0 + l16
    float* g = ws_gram + (size_t)bid * GRAM_ELEMS;
    #pragma unroll
    for (int n = 0; n < 8; ++n)
        #pragma unroll
        for (int i = 0; i < 8; ++i) {
            int row = m0 + i + (lhalf ? 8 : 0);
            int col = (n << 4) + l16;
            g[row * 128 + col] = acc[n][i];
        }
}

// ------------------------------------------------------------------
// Per MI unit: sigma^2 from 10-smallest sqrt distances, then kx/ky/kxy
// ------------------------------------------------------------------
__global__ __launch_bounds__(128)
void mi_prep_kernel(const float* __restrict__ ws_gram, float* __restrict__ ws_kmat)
{
    __shared__ float diag[128];
    __shared__ float red[128];
    __shared__ float sig2[2];
    __shared__ float diagx[128], diagy[128];
    int u = blockIdx.x;
    int t = threadIdx.x;
    const float* Gx = ws_gram + (size_t)(u * 2 + 0) * GRAM_ELEMS;
    const float* Gm = ws_gram + (size_t)(u * 2 + 1) * GRAM_ELEMS;
    float* Kx  = ws_kmat + (size_t)(u * 3 + 0) * GRAM_ELEMS;
    float* Ky  = ws_kmat + (size_t)(u * 3 + 1) * GRAM_ELEMS;
    float* Kxy = ws_kmat + (size_t)(u * 3 + 2) * GRAM_ELEMS;

    for (int pass = 0; pass < 2; ++pass) {
        const float* G = pass ? Gm : Gx;
        diag[t] = G[t * 128 + t];
        __syncthreads();
        float best[10];
        #pragma unroll
        for (int k = 0; k < 10; ++k) best[k] = 1e30f;
        float dt = diag[t];
        for (int j = 0; j < 128; ++j) {
            float dsq = fmaxf(dt + diag[j] - 2.f * G[t * 128 + j], 0.f);
            float dd = sqrtf(dsq);
            if (dd < best[9]) {
                best[9] = dd;
                #pragma unroll
                for (int k = 9; k > 0; --k)
                    if (best[k] < best[k - 1]) { float tmp = best[k]; best[k] = best[k - 1]; best[k - 1] = tmp; }
            }
        }
        float s = 0.f;
        #pragma unroll
        for (int k = 0; k < 10; ++k) s += best[k];
        red[t] = s;
        __syncthreads();
        for (int off = 64; off > 0; off >>= 1) {
            if (t < off) red[t] += red[t + off];
            __syncthreads();
        }
        if (t == 0) {
            float sm = red[0] / 1280.f;
            sm = fmaxf(sm, 0.01f);
            sig2[pass] = sm * sm;
        }
        __syncthreads();
    }

    diagx[t] = Gx[t * 128 + t];
    diagy[t] = Gm[t * 128 + t];
    __syncthreads();
    float sx2 = sig2[0], sy2 = sig2[1];
    float dxt = diagx[t], dyt = diagy[t];
    for (int j = 0; j < 128; ++j) {
        float dsx = fmaxf(dxt + diagx[j] - 2.f * Gx[t * 128 + j], 0.f);
        float dsy = fmaxf(dyt + diagy[j] - 2.f * Gm[t * 128 + j], 0.f);
        float kx = expf(-dsx / sx2);
        float ky = expf(-dsy / sy2);
        Kx[t * 128 + j]  = kx;
        Ky[t * 128 + j]  = ky;
        Kxy[t * 128 + j] = kx * ky;
    }
}

// ------------------------------------------------------------------
// Parallel cyclic Jacobi eigensolver + Renyi entropy (alpha = 1.01)
// one block per (unit, gram-of-3); matrix lives entirely in LDS
// ------------------------------------------------------------------
#define NSWEEP 8
__global__ __launch_bounds__(128)
void jacobi_entropy_kernel(const float* __restrict__ ws_kmat, float* __restrict__ ws_ent)
{
    __shared__ float A[128][129];
    __shared__ int   perm[128];
    __shared__ int   pp[64], qq[64];
    __shared__ float cv[64], sv[64];
    __shared__ float red[128];
    __shared__ float invtr_sh;
    int bid = blockIdx.x;   // 0..143
    int t = threadIdx.x;
    const float* K = ws_kmat + (size_t)bid * GRAM_ELEMS;

    for (int i = t; i < GRAM_ELEMS; i += 128) A[i >> 7][i & 127] = K[i];
    __syncthreads();
    red[t] = A[t][t];
    __syncthreads();
    for (int off = 64; off > 0; off >>= 1) {
        if (t < off) red[t] += red[t + off];
        __syncthreads();
    }
    if (t == 0) invtr_sh = 1.f / red[0];
    __syncthreads();
    float invtr = invtr_sh;
    for (int i = t; i < GRAM_ELEMS; i += 128) A[i >> 7][i & 127] *= invtr;
    perm[t] = t;
    __syncthreads();

    for (int sweep = 0; sweep < NSWEEP; ++sweep) {
        for (int step = 0; step < 127; ++step) {
            if (t < 64) {
                int p = perm[t], q = perm[127 - t];
                pp[t] = p; qq[t] = q;
                float apq = A[p][q];
                float c = 1.f, s = 0.f;
                if (fabsf(apq) > 1e-24f) {
                    float theta = (A[q][q] - A[p][p]) / (2.f * apq);
                    float tt = ((theta >= 0.f) ? 1.f : -1.f) /
                               (fabsf(theta) + sqrtf(theta * theta + 1.f));
                    c = 1.f / sqrtf(1.f + tt * tt);
                    s = tt * c;
                }
                cv[t] = c; sv[t] = s;
            }
            __syncthreads();
            // column rotations: thread owns row t  (banks conflict-free via pad 129)
            for (int j = 0; j < 64; ++j) {
                float c = cv[j], s = sv[j]; int p = pp[j], q = qq[j];
                float xv = A[t][p], yv = A[t][q];
                A[t][p] = c * xv - s * yv;
                A[t][q] = s * xv + c * yv;
            }
            __syncthreads();
            // row rotations: thread owns column t
            for (int j = 0; j < 64; ++j) {
                float c = cv[j], s = sv[j]; int p = pp[j], q = qq[j];
                float xv = A[p][t], yv = A[q][t];
                A[p][t] = c * xv - s * yv;
                A[q][t] = s * xv + c * yv;
            }
            __syncthreads();
            // round-robin rotate (element 0 fixed)
            int oldv = perm[t];
            __syncthreads();
            int dst = (t == 0) ? 0 : ((t == 127) ? 1 : t + 1);
            perm[dst] = oldv;
            __syncthreads();
        }
    }

    red[t] = powf(fabsf(A[t][t]), 1.01f);
    __syncthreads();
    for (int off = 64; off > 0; off >>= 1) {
        if (t < off) red[t] += red[t + off];
        __syncthreads();
    }
    if (t == 0) ws_ent[bid] = log2f(red[0]) * (1.f / (1.f - 1.01f));
}

__global__ __launch_bounds__(64)
void finalize_xt_kernel(const float* __restrict__ ws_ent, float* __restrict__ out_xt)
{
    int t = threadIdx.x;
    if (t < NUNIT)
        out_xt[t] = (ws_ent[t * 3 + 0] + ws_ent[t * 3 + 1] - ws_ent[t * 3 + 2]); // BETA = 1
}

// ------------------------------------------------------------------
// Decoder conv (stride 4, 4x4, 3->16) + ReLU + spatial mean, per batch
// ------------------------------------------------------------------
__global__ __launch_bounds__(256)
void dec_pool_kernel(const float* __restrict__ masked, const float* __restrict__ dec_w,
                     const float* __restrict__ dec_b, float* __restrict__ hpool)
{
    __shared__ float wsm[768];
    __shared__ float red[256];
    int b = blockIdx.x;
    int tid = threadIdx.x;
    for (int i = tid; i < 768; i += 256) wsm[i] = dec_w[i];
    __syncthreads();

    float acc[16];
    #pragma unroll
    for (int oc = 0; oc < 16; ++oc) acc[oc] = 0.f;

    for (int p = tid; p < 4096; p += 256) {
        int oy = p >> 6, ox = p & 63;
        float in[48];
        #pragma unroll
        for (int ic = 0; ic < 3; ++ic)
            #pragma unroll
            for (int ky = 0; ky < 4; ++ky)
                #pragma unroll
                for (int kx = 0; kx < 4; ++kx)
                    in[(ic * 4 + ky) * 4 + kx] =
                        masked[(((size_t)(b * 3 + ic) * 256 + oy * 4 + ky) * 256) + ox * 4 + kx];
        #pragma unroll
        for (int oc = 0; oc < 16; ++oc) {
            float s = dec_b[oc];
            #pragma unroll
            for (int r = 0; r < 48; ++r) s += in[r] * wsm[oc * 48 + r];
            acc[oc] += fmaxf(s, 0.f);
        }
    }
    for (int oc = 0; oc < 16; ++oc) {
        red[tid] = acc[oc];
        __syncthreads();
        for (int off = 128; off > 0; off >>= 1) {
            if (tid < off) red[tid] += red[tid + off];
            __syncthreads();
        }
        if (tid == 0) hpool[b * 16 + oc] = red[0] / 4096.f;
        __syncthreads();
    }
}

__global__ __launch_bounds__(128)
void fc_kernel(const float* __restrict__ hpool, const float* __restrict__ fc_w,
               const float* __restrict__ fc_b, float* __restrict__ out_logit)
{
    int b = threadIdx.x;
    for (int cls = 0; cls < 10; ++cls) {
        float s = fc_b[cls];
        #pragma unroll
        for (int k = 0; k < 16; ++k) s += hpool[b * 16 + k] * fc_w[cls * 16 + k];
        out_logit[b * 10 + cls] = s;
    }
}

// ------------------------------------------------------------------
extern "C" void kernel_launch(void* const* d_in, const int* in_sizes, int n_in,
                              void* d_out, int out_size, void* d_ws, size_t ws_size,
                              hipStream_t stream)
{
    (void)in_sizes; (void)n_in; (void)out_size; (void)ws_size;
    const float* x      = (const float*)d_in[0];
    const float* prior  = (const float*)d_in[1];
    const float* enc_w  = (const float*)d_in[2];
    const float* enc_b  = (const float*)d_in[3];
    const float* dec_w  = (const float*)d_in[4];
    const float* dec_b  = (const float*)d_in[5];
    const float* fc_w   = (const float*)d_in[6];
    const float* fc_b   = (const float*)d_in[7];
    float* out = (float*)d_out;
    float* ws  = (float*)d_ws;

    float* zsmall = ws + WS_ZSMALL;
    float* masked = ws + WS_MASKED;
    float* gram   = ws + WS_GRAM;
    float* kmat   = ws + WS_KMAT;
    float* ent    = ws + WS_ENT;
    float* hpool  = ws + WS_HPOOL;

    enc_kernel<<<6144, 256, 0, stream>>>(x, prior, enc_w, enc_b, zsmall, out + OUT_MP);
    upsample_mask_kernel<<<98304, 256, 0, stream>>>(x, zsmall, out + OUT_Z, masked);
    gram_kernel<<<96, 256, 0, stream>>>(x, masked, gram);
    mi_prep_kernel<<<NUNIT, 128, 0, stream>>>(gram, kmat);
    jacobi_entropy_kernel<<<144, 128, 0, stream>>>(kmat, ent);
    finalize_xt_kernel<<<1, 64, 0, stream>>>(ent, out + OUT_XT);
    dec_pool_kernel<<<128, 256, 0, stream>>>(masked, dec_w, dec_b, hpool);
    fc_kernel<<<1, 128, 0, stream>>>(hpool, fc_w, fc_b, out + OUT_LOGIT);
}